// Net_26740466385314
// MI455X (gfx1250) — compile-verified
//
#include <hip/hip_runtime.h>

#define N_NODES 50000
#define IN_F    96
#define H_F     256
#define OUT_F   40
#define N_EDGES 800000

typedef __attribute__((ext_vector_type(2))) float v2f;
typedef __attribute__((ext_vector_type(8))) float v8f;

// ---------------- utility kernels ----------------

__global__ void zero_f32(float* p, int n) {
    int i = blockIdx.x * blockDim.x + threadIdx.x;
    if (i < n) p[i] = 0.0f;
}

__global__ void init_out_bias(float* out, const float* __restrict__ b2, int n) {
    int i = blockIdx.x * blockDim.x + threadIdx.x;
    if (i < n) out[i] = b2[i % OUT_F];
}

__global__ void degree_kernel(const int* __restrict__ src, const int* __restrict__ dst,
                              float* deg_out, float* deg_in, int e) {
    int i = blockIdx.x * blockDim.x + threadIdx.x;
    if (i < e) {
        atomicAdd(&deg_out[src[i]], 1.0f);
        atomicAdd(&deg_in[dst[i]], 1.0f);
    }
}

__global__ void norm_kernel(float* deg_out, float* deg_in, int n) {
    int i = blockIdx.x * blockDim.x + threadIdx.x;
    if (i < n) {
        deg_out[i] = rsqrtf(fmaxf(deg_out[i], 1.0f));  // becomes norm_out
        deg_in[i]  = rsqrtf(fmaxf(deg_in[i],  1.0f));  // becomes norm_in
    }
}

// layer-1 aggregation: agg[dst] += x[src] * norm_out[src], 4 feats per thread
__global__ void scatter1(const float* __restrict__ x, const int* __restrict__ src,
                         const int* __restrict__ dst, const float* __restrict__ norm_out,
                         float* agg) {
    int i = blockIdx.x * blockDim.x + threadIdx.x;
    if (i >= N_EDGES * (IN_F / 4)) return;
    int e  = i / (IN_F / 4);
    int fg = i % (IN_F / 4);
    int s = src[e], d = dst[e];
    float ns = norm_out[s];
    float4 v = reinterpret_cast<const float4*>(x + (size_t)s * IN_F)[fg];
    float* a = agg + (size_t)d * IN_F + fg * 4;
    atomicAdd(a + 0, v.x * ns);
    atomicAdd(a + 1, v.y * ns);
    atomicAdd(a + 2, v.z * ns);
    atomicAdd(a + 3, v.w * ns);
}

// layer-2 scatter: out[dst] += t2[src] * norm_in[dst], 4 feats per thread
__global__ void scatter2(const float* __restrict__ t2, const int* __restrict__ src,
                         const int* __restrict__ dst, const float* __restrict__ norm_in,
                         float* out) {
    int i = blockIdx.x * blockDim.x + threadIdx.x;
    if (i >= N_EDGES * (OUT_F / 4)) return;
    int e  = i / (OUT_F / 4);
    int fg = i % (OUT_F / 4);
    int s = src[e], d = dst[e];
    float nd = norm_in[d];
    float4 v = reinterpret_cast<const float4*>(t2 + (size_t)s * OUT_F)[fg];
    float* o = out + (size_t)d * OUT_F + fg * 4;
    atomicAdd(o + 0, v.x * nd);
    atomicAdd(o + 1, v.y * nd);
    atomicAdd(o + 2, v.z * nd);
    atomicAdd(o + 3, v.w * nd);
}

// ---------------- WMMA GEMMs (full-precision f32, 16x16x4) ----------------
// h1[m, 0:256] = relu( (agg[m,:] * norm_in[m]) @ W1 + b1 ), 16 rows per block.
// A-matrix 16x4 f32 layout: lane = 16*kh + row holds {A[row][2kh], A[row][2kh+1]}.
// B-matrix 4x16 f32 layout: lane = 16*kh + n  holds {B[2kh][n],  B[2kh+1][n]}.
// C/D 16x16: lane = 16*g + n, VGPR v holds C[v + 8*g][n].
__global__ void __launch_bounds__(256) gemm1(const float* __restrict__ agg,
                                             const float* __restrict__ norm_in,
                                             const float* __restrict__ W1,
                                             const float* __restrict__ b1,
                                             float* __restrict__ h1) {
    __shared__ float As[16 * IN_F];
    int mbase = blockIdx.x * 16;
    for (int i = threadIdx.x; i < 16 * IN_F; i += 256) {
        int row = i / IN_F, k = i % IN_F;
        As[i] = agg[(size_t)(mbase + row) * IN_F + k] * norm_in[mbase + row];
    }
    __syncthreads();

    int lane = threadIdx.x & 31;
    int wave = threadIdx.x >> 5;
    int row  = lane & 15;     // A-row / B-col within tile
    int kh   = lane >> 4;     // K-half selector

    for (int t = 0; t < 2; ++t) {
        int colbase = (wave * 2 + t) * 16;     // 8 waves x 2 tiles = 256 cols
        v8f c = {};
        for (int k0 = 0; k0 < IN_F; k0 += 4) {
            int k = k0 + 2 * kh;
            v2f a = { As[row * IN_F + k], As[row * IN_F + k + 1] };
            v2f b = { W1[(size_t)k * H_F + colbase + row],
                      W1[(size_t)(k + 1) * H_F + colbase + row] };
            c = __builtin_amdgcn_wmma_f32_16x16x4_f32(false, a, false, b,
                                                      (short)0, c, false, false);
        }
        int n = colbase + row;
        float bias = b1[n];
        for (int v = 0; v < 8; ++v) {
            int m = v + 8 * kh;
            float val = c[v] + bias;
            h1[(size_t)(mbase + m) * H_F + n] = fmaxf(val, 0.0f);
        }
    }
}

// t2[m, 0:40] = (h1[m,:] * norm_out[m]) @ W2 ; 16 rows per block, 3 compute waves.
__global__ void __launch_bounds__(128) gemm2(const float* __restrict__ h1,
                                             const float* __restrict__ norm_out,
                                             const float* __restrict__ W2,
                                             float* __restrict__ t2) {
    __shared__ float As[16 * H_F];
    int mbase = blockIdx.x * 16;
    for (int i = threadIdx.x; i < 16 * H_F; i += 128) {
        int row = i >> 8, k = i & 255;
        As[i] = h1[(size_t)(mbase + row) * H_F + k] * norm_out[mbase + row];
    }
    __syncthreads();

    int wave = threadIdx.x >> 5;
    if (wave < 3) {                       // whole-wave branch: EXEC all-1 inside
        int lane = threadIdx.x & 31;
        int row  = lane & 15;
        int kh   = lane >> 4;
        int colbase = wave * 16;
        int n = colbase + row;
        v8f c = {};
        for (int k0 = 0; k0 < H_F; k0 += 4) {
            int k = k0 + 2 * kh;
            v2f a = { As[row * H_F + k], As[row * H_F + k + 1] };
            v2f b = { (n < OUT_F) ? W2[(size_t)k * OUT_F + n] : 0.0f,
                      (n < OUT_F) ? W2[(size_t)(k + 1) * OUT_F + n] : 0.0f };
            c = __builtin_amdgcn_wmma_f32_16x16x4_f32(false, a, false, b,
                                                      (short)0, c, false, false);
        }
        if (n < OUT_F) {
            for (int v = 0; v < 8; ++v) {
                int m = v + 8 * kh;
                t2[(size_t)(mbase + m) * OUT_F + n] = c[v];
            }
        }
    }
}

// ---------------- launch ----------------

extern "C" void kernel_launch(void* const* d_in, const int* in_sizes, int n_in,
                              void* d_out, int out_size, void* d_ws, size_t ws_size,
                              hipStream_t stream) {
    const float* x   = (const float*)d_in[0];
    const int*   src = (const int*)  d_in[1];
    const int*   dst = (const int*)  d_in[2];
    const float* W1  = (const float*)d_in[3];
    const float* b1  = (const float*)d_in[4];
    const float* W2  = (const float*)d_in[5];
    const float* b2  = (const float*)d_in[6];
    float* out = (float*)d_out;

    float* ws       = (float*)d_ws;
    float* norm_out = ws;                               // [N]
    float* norm_in  = norm_out + N_NODES;               // [N]
    float* agg1     = norm_in + N_NODES;                // [N, 96]
    float* h1       = agg1 + (size_t)N_NODES * IN_F;    // [N, 256]
    float* t2       = h1 + (size_t)N_NODES * H_F;       // [N, 40]

    const int TB = 256;
    // 1) degrees -> norms (deg arrays become norm arrays in place)
    zero_f32<<<(2 * N_NODES + TB - 1) / TB, TB, 0, stream>>>(norm_out, 2 * N_NODES);
    degree_kernel<<<(N_EDGES + TB - 1) / TB, TB, 0, stream>>>(src, dst, norm_out, norm_in, N_EDGES);
    norm_kernel<<<(N_NODES + TB - 1) / TB, TB, 0, stream>>>(norm_out, norm_in, N_NODES);

    // 2) layer-1 aggregation
    zero_f32<<<(N_NODES * IN_F + TB - 1) / TB, TB, 0, stream>>>(agg1, N_NODES * IN_F);
    scatter1<<<(N_EDGES * (IN_F / 4) + TB - 1) / TB, TB, 0, stream>>>(x, src, dst, norm_out, agg1);

    // 3) layer-1 GEMM + bias + relu (3125 M-tiles of 16)
    gemm1<<<N_NODES / 16, 256, 0, stream>>>(agg1, norm_in, W1, b1, h1);

    // 4) layer-2 GEMM (project before aggregate)
    gemm2<<<N_NODES / 16, 128, 0, stream>>>(h1, norm_out, W2, t2);

    // 5) layer-2 aggregation into output (init with bias)
    init_out_bias<<<(N_NODES * OUT_F + TB - 1) / TB, TB, 0, stream>>>(out, b2, N_NODES * OUT_F);
    scatter2<<<(N_EDGES * (OUT_F / 4) + TB - 1) / TB, TB, 0, stream>>>(t2, src, dst, norm_in, out);
}